// fTN_NNiso_Model_1391569404360
// MI455X (gfx1250) — compile-verified
//
#include <hip/hip_runtime.h>
#include <hip/hip_bf16.h>

typedef float v2f __attribute__((ext_vector_type(2)));
typedef float v8f __attribute__((ext_vector_type(8)));

#define LX 6
#define LY 6
#define DDIM 4
#define CHI 16
#define HDIM 64
#define NSITES 36
#define PROJ 51200
#define NTILE (PROJ / 16)   // 3200
#define BATCH 1024
#define ETA 0.001f

// peps flat index: (Lx, Ly, D, D, D, D, PHYS) row-major
__device__ __forceinline__ int pidx(int r, int c, int l, int rr, int a, int b, int xi) {
    return ((((((r * LY + c) * DDIM + l) * DDIM + rr) * DDIM + a) * DDIM + b) * 2 + xi);
}

__device__ __forceinline__ v8f wmma4(v2f a, v2f b, v8f c) {
    // V_WMMA_F32_16X16X4_F32: D(16x16 f32) = A(16x4 f32) x B(4x16 f32) + C
    return __builtin_amdgcn_wmma_f32_16x16x4_f32(false, a, false, b, (short)0, c, false, false);
}

// ---------------- Kernel 1: h = relu(x @ W1 + b1) for all 1024 samples ----------------
__global__ void nn_hidden_kernel(const int* __restrict__ x, const float* __restrict__ W1,
                                 const float* __restrict__ b1, float* __restrict__ Hws) {
    const int s = blockIdx.x;     // sample
    const int j = threadIdx.x;    // hidden unit (64)
    float acc = b1[j];
    #pragma unroll
    for (int i = 0; i < NSITES; ++i)
        acc += (float)x[s * NSITES + i] * W1[i * HDIM + j];
    Hws[s * HDIM + j] = fmaxf(acc, 0.0f);
}

// ---------------- Kernel 2: VEC = base_proj + ETA*(H @ W2 + b2), one sample chunk ----------------
// One wave per 16x16 output tile; K=64 via 16 chained v_wmma_f32_16x16x4_f32.
__global__ __launch_bounds__(128) void vec_gemm_kernel(const float* __restrict__ Hc,
                                                       const float* __restrict__ W2,
                                                       const float* __restrict__ b2,
                                                       const float* __restrict__ bp,
                                                       float* __restrict__ VEC) {
    const int lane = threadIdx.x & 31;
    const int wave = threadIdx.x >> 5;
    const int tile = blockIdx.x * 4 + wave;     // (cs/16 sample-tiles) x 3200 n-tiles
    const int nTile = tile % NTILE;
    const int sTile = tile / NTILE;
    const int half = lane >> 4;
    const int nl = lane & 15;
    const int n = nTile * 16 + nl;

    v8f acc = {0.f, 0.f, 0.f, 0.f, 0.f, 0.f, 0.f, 0.f};
    const float* hrow = Hc + (sTile * 16 + nl) * HDIM;
    #pragma unroll
    for (int kk = 0; kk < 16; ++kk) {
        const int k0 = 4 * kk + 2 * half;
        v2f a, b;
        a.x = hrow[k0];
        a.y = hrow[k0 + 1];
        b.x = W2[(size_t)k0 * PROJ + n];
        b.y = W2[(size_t)(k0 + 1) * PROJ + n];
        acc = wmma4(a, b, acc);
    }
    const float bpn = bp[n];
    const float bbn = b2[n];
    #pragma unroll
    for (int i = 0; i < 8; ++i) {
        const int m = sTile * 16 + i + 8 * half;
        VEC[(size_t)m * PROJ + n] = bpn + ETA * (acc[i] + bbn);
    }
}

// ---------------- Kernel 3: PEPS column-sweep contraction, one block per sample ----------------
// 128 threads = 4 waves. P-compress (64x64)@(64x16) and Q-compress (16x64)@(64x16) use WMMA.
__global__ __launch_bounds__(128) void tn_contract_kernel(const float* __restrict__ VEC,
                                                          const float* __restrict__ peps,
                                                          const int* __restrict__ xc,
                                                          float* __restrict__ outc) {
    __shared__ float Ms[LX][CHI][CHI][4];   // 24 KB  per-row compressed state (zero padded)
    __shared__ float T[64][64];             // 16 KB  grown tensor, one r' slice at a time
    __shared__ float tmp[4][64][16];        // 16 KB  after P-compression (all r' slices)
    __shared__ float Ae[4][4][4][4];        //  1 KB  site tensor (l, r', a, b), zero padded
    __shared__ float vv[2][16];

    const int sL = blockIdx.x;              // sample within chunk
    const int tid = threadIdx.x;
    const int lane = tid & 31;
    const int wave = tid >> 5;
    const int half = lane >> 4;
    const int nl = lane & 15;
    const size_t sBase = (size_t)sL * PROJ;

    // ---- init Ms from column 0 ----
    for (int idx = tid; idx < LX * CHI * CHI * 4; idx += 128) {
        const int r = idx >> 10;
        const int rem = idx & 1023;
        const int U = rem >> 6;
        const int Dd = (rem >> 2) & 15;
        const int rr = rem & 3;
        const int asz = (r == 0) ? 1 : 4;
        const int bsz = (r == LX - 1) ? 1 : 4;
        float v = 0.f;
        if (U < asz && Dd < bsz) {
            const int xi = xc[sL * NSITES + r * LY + 0];
            v = peps[pidx(r, 0, 0, rr, U, Dd, xi)];
        }
        Ms[r][U][Dd][rr] = v;
    }
    __syncthreads();

    // ---- column sweep ----
    for (int c = 1; c < LY; ++c) {
        for (int r = 0; r < LX; ++r) {
            const size_t pB = sBase + (size_t)(((c - 1) * 5 + r) * 2 + 0) * 1024;        // P(64,16)
            const size_t qB = sBase + (size_t)(((c - 1) * 5 + (r - 1)) * 2 + 1) * 1024;  // Q(16,64)

            // prefetch the P/Q bond matrices while the grow step runs (global_prefetch_b8)
            if (r < LX - 1) __builtin_prefetch(&VEC[pB + (size_t)tid * 8], 0, 1);
            if (r > 0)      __builtin_prefetch(&VEC[qB + (size_t)tid * 8], 0, 1);

            // load zero-padded site tensor Ae[l][rr][a][b]
            const int xi = xc[sL * NSITES + r * LY + c];
            for (int idx = tid; idx < 256; idx += 128) {
                const int l = idx >> 6;
                const int rr = (idx >> 4) & 3;
                const int a = (idx >> 2) & 3;
                const int b = idx & 3;
                const bool valid = (c < LY - 1 || rr == 0) && (r > 0 || a == 0) && (r < LX - 1 || b == 0);
                Ae[l][rr][a][b] = valid ? peps[pidx(r, c, l, rr, a, b, xi)] : 0.f;
            }
            __syncthreads();

            for (int rr = 0; rr < 4; ++rr) {
                // grow: T[u][d] = sum_l Ms[r][U][Dd][l] * Ae[l][rr][a][b]
                for (int idx = tid; idx < 4096; idx += 128) {
                    const int u = idx >> 6, d = idx & 63;
                    const int U = u >> 2, a = u & 3, Dd = d >> 2, b = d & 3;
                    float acc = 0.f;
                    #pragma unroll
                    for (int l = 0; l < 4; ++l)
                        acc += Ms[r][U][Dd][l] * Ae[l][rr][a][b];
                    T[u][d] = acc;
                }
                __syncthreads();

                if (r < LX - 1) {
                    // P-compress with WMMA: tmp[rr] = T(64x64) @ P(64x16); wave = m-tile
                    v8f acc = {0.f, 0.f, 0.f, 0.f, 0.f, 0.f, 0.f, 0.f};
                    #pragma unroll
                    for (int kk = 0; kk < 16; ++kk) {
                        const int k0 = 4 * kk + 2 * half;
                        v2f a, b;
                        a.x = T[wave * 16 + nl][k0];
                        a.y = T[wave * 16 + nl][k0 + 1];
                        b.x = VEC[pB + k0 * 16 + nl];
                        b.y = VEC[pB + (k0 + 1) * 16 + nl];
                        acc = wmma4(a, b, acc);
                    }
                    #pragma unroll
                    for (int i = 0; i < 8; ++i)
                        tmp[rr][wave * 16 + i + 8 * half][nl] = acc[i];
                } else {
                    // bottom row: no P; only d<16 can be nonzero
                    for (int idx = tid; idx < 1024; idx += 128)
                        tmp[rr][idx >> 4][idx & 15] = T[idx >> 4][idx & 15];
                }
                __syncthreads();
            }

            // update Ms[r]
            if (r > 0) {
                // Q-compress with WMMA: Ms[r][.,.,rr] = Q(16x64) @ tmp[rr](64x16); wave = rr
                const int rr = wave;
                v8f acc = {0.f, 0.f, 0.f, 0.f, 0.f, 0.f, 0.f, 0.f};
                #pragma unroll
                for (int kk = 0; kk < 16; ++kk) {
                    const int k0 = 4 * kk + 2 * half;
                    v2f a, b;
                    a.x = VEC[qB + nl * 64 + k0];
                    a.y = VEC[qB + nl * 64 + k0 + 1];
                    b.x = tmp[rr][k0][nl];
                    b.y = tmp[rr][k0 + 1][nl];
                    acc = wmma4(a, b, acc);
                }
                #pragma unroll
                for (int i = 0; i < 8; ++i)
                    Ms[r][i + 8 * half][nl][rr] = acc[i];
            } else {
                // top row: no Q; only u<16 rows of tmp can be nonzero
                for (int idx = tid; idx < 1024; idx += 128) {
                    const int U = idx >> 6;
                    const int cc = (idx >> 2) & 15;
                    const int rr = idx & 3;
                    Ms[0][U][cc][rr] = tmp[rr][U][cc];
                }
            }
            __syncthreads();
        }
    }

    // ---- final boundary vector reduce ----
    if (tid < 16) vv[0][tid] = Ms[0][0][tid][0];
    __syncthreads();
    for (int r = 1; r < LX - 1; ++r) {
        if (tid < 16) {
            float acc = 0.f;
            #pragma unroll
            for (int i = 0; i < 16; ++i)
                acc += vv[(r - 1) & 1][i] * Ms[r][i][tid][0];
            vv[r & 1][tid] = acc;
        }
        __syncthreads();
    }
    if (tid == 0) {
        float acc = 0.f;
        #pragma unroll
        for (int i = 0; i < 16; ++i)
            acc += vv[(LX - 2) & 1][i] * Ms[LX - 1][i][0][0];
        outc[sL] = acc;
    }
}

extern "C" void kernel_launch(void* const* d_in, const int* in_sizes, int n_in,
                              void* d_out, int out_size, void* d_ws, size_t ws_size,
                              hipStream_t stream) {
    const int*   x         = (const int*)d_in[0];
    const float* peps      = (const float*)d_in[1];
    const float* base_proj = (const float*)d_in[2];
    const float* W1        = (const float*)d_in[3];
    const float* b1        = (const float*)d_in[4];
    const float* W2        = (const float*)d_in[5];
    const float* b2        = (const float*)d_in[6];
    float* out = (float*)d_out;

    float* Hws = (float*)d_ws;                 // 1024*64 f32 = 256 KB
    float* VEC = Hws + (size_t)BATCH * HDIM;   // chunkS*51200 f32 (reused per chunk)

    // Deterministic chunk sizing from ws_size: largest multiple-of-16 sample chunk
    // whose VEC slab fits the workspace (maximizes TN-phase occupancy).
    const size_t hBytes = (size_t)BATCH * HDIM * sizeof(float);
    size_t avail = (ws_size > hBytes) ? (ws_size - hBytes) : 0;
    int chunkS = (int)(avail / ((size_t)PROJ * sizeof(float)));
    chunkS = (chunkS / 16) * 16;
    if (chunkS > BATCH) chunkS = BATCH;
    if (chunkS < 16) chunkS = 16;   // minimal fallback

    nn_hidden_kernel<<<BATCH, HDIM, 0, stream>>>(x, W1, b1, Hws);

    for (int s0 = 0; s0 < BATCH; s0 += chunkS) {
        const int cs = (BATCH - s0 < chunkS) ? (BATCH - s0) : chunkS;   // multiple of 16
        // (cs/16 sample-tiles) * 3200 n-tiles wave-tiles, 4 waves/block
        vec_gemm_kernel<<<(cs / 16) * (NTILE / 4), 128, 0, stream>>>(
            Hws + (size_t)s0 * HDIM, W2, b2, base_proj, VEC);
        tn_contract_kernel<<<cs, 128, 0, stream>>>(
            VEC, peps, x + (size_t)s0 * NSITES, out + s0);
    }
}